// HashEmbedding_32401233281223
// MI455X (gfx1250) — compile-verified
//
#include <hip/hip_runtime.h>
#include <hip/hip_bf16.h>

// HashEmbedding on MI455X (gfx1250): pure gather + streamed-store workload.
// Memory-bound: ~216MB NT output stores + L2-resident gathers (W=25.6MB,
// P/ht=16MB all fit in 192MB L2). WMMA does not apply (per-row gathered
// operands, K=2). CDNA5 paths used: global_load_async_to_lds_b32 (+
// s_wait_asynccnt) for id staging, global_prefetch_b8 for the dependent
// gather chain, non-temporal b64 stores for the output stream.

typedef float v4f __attribute__((ext_vector_type(4)));
typedef float v2f __attribute__((ext_vector_type(2)));

#define EMBED   64
#define OUTW    66   // 64 emb + 2 pvals
#define NHASH   2
#define TPB     256  // 8 wave32s; 32 words per wave; 256 words per block

__global__ __launch_bounds__(TPB, 4)
void hash_embed_gfx1250(const int*   __restrict__ words,   // [B*L]
                        const int*   __restrict__ ht,      // [NUM_WORDS,2]
                        const float* __restrict__ W,       // [NUM_BUCKETS,64]
                        const float* __restrict__ P,       // [NUM_WORDS,2]
                        float*       __restrict__ out,     // [B*L,66]
                        int total_words) {
    __shared__ int ids[TPB];

    const int tid  = threadIdx.x;
    const int lane = tid & 31;
    const int wave = tid >> 5;
    const int half = lane >> 4;   // which of the 2 words this half-wave owns
    const int sub  = lane & 15;   // float4 slice index within the 64-wide row

    const int blockBase = blockIdx.x * TPB;

    // ---- Stage this block's 256 word ids into LDS via the CDNA5 async path.
    // One b32 per lane; each wave only consumes its own 32-id slice, so a
    // per-wave s_wait_asynccnt is sufficient (no workgroup barrier needed).
    {
        int gi = blockBase + tid;
        if (gi >= total_words) gi = total_words - 1;      // clamp, avoid OOB
        unsigned            lds_off = (unsigned)(size_t)(&ids[tid]);
        unsigned long long  gaddr   = (unsigned long long)(size_t)(words + gi);
        asm volatile("global_load_async_to_lds_b32 %0, %1, off"
                     :: "v"(lds_off), "v"(gaddr) : "memory");
        asm volatile("s_wait_asynccnt 0" ::: "memory");
    }

    const int waveBase = wave * 32;

    #pragma unroll 4
    for (int i = 0; i < 16; ++i) {
        const int widx = blockBase + waveBase + 2 * i + half;  // global word
        if (widx >= total_words) continue;

        const int word = ids[waveBase + 2 * i + half];

        // Prefetch next pair's hash/P cachelines: hides the id->hash->W chain.
        if (i < 15) {
            const int wnext = ids[waveBase + 2 * i + 2 + half];
            __builtin_prefetch(ht + 2 * (size_t)wnext, 0, 3);
            __builtin_prefetch(P  + 2 * (size_t)wnext, 0, 3);
        }

        // Uniform per half-wave (broadcast through L0).
        const int   h0 = ht[2 * (size_t)word + 0];
        const int   h1 = ht[2 * (size_t)word + 1];
        const float pw0 = P[2 * (size_t)word + 0];
        const float pw1 = P[2 * (size_t)word + 1];

        // Coalesced b128 gathers of the two W rows (rows are 256B aligned).
        const v4f w0 = *(const v4f*)(W + (size_t)h0 * EMBED + sub * 4);
        const v4f w1 = *(const v4f*)(W + (size_t)h1 * EMBED + sub * 4);

        v4f r = w0 * pw0 + w1 * pw1;

        // Output row is only 8B aligned (66 floats/row) -> NT float2 stores.
        float* orow = out + (size_t)widx * OUTW;
        v2f lo = { r.x, r.y };
        v2f hi = { r.z, r.w };
        __builtin_nontemporal_store(lo, (v2f*)(orow + sub * 4));
        __builtin_nontemporal_store(hi, (v2f*)(orow + sub * 4 + 2));

        // pvals[b,l,h] = P[hash_h, h]; one lane per word writes the tail pair.
        if (sub == 0) {
            v2f pv = { P[2 * (size_t)h0 + 0], P[2 * (size_t)h1 + 1] };
            __builtin_nontemporal_store(pv, (v2f*)(orow + EMBED));
        }
    }
}

extern "C" void kernel_launch(void* const* d_in, const int* in_sizes, int n_in,
                              void* d_out, int out_size, void* d_ws, size_t ws_size,
                              hipStream_t stream) {
    (void)n_in; (void)out_size; (void)d_ws; (void)ws_size;
    const int*   words = (const int*)  d_in[0];   // [B*L]
    const int*   ht    = (const int*)  d_in[1];   // [NUM_WORDS, 2]
    const float* W     = (const float*)d_in[2];   // [NUM_BUCKETS, 64]
    const float* P     = (const float*)d_in[3];   // [NUM_WORDS, 2]
    float*       out   = (float*)      d_out;     // [B*L, 66]

    const int total_words = in_sizes[0];          // 16384 * 50 = 819200
    const int blocks = (total_words + TPB - 1) / TPB;

    hash_embed_gfx1250<<<blocks, TPB, 0, stream>>>(words, ht, W, P, out,
                                                   total_words);
}